// TCNNHashPositionalEncoder_56581899157969
// MI455X (gfx1250) — compile-verified
//
#include <hip/hip_runtime.h>
#include <cstdint>

#define NLEV   16
#define LOG2T  19
#define TMASK  ((1u << LOG2T) - 1u)
#define PRIME1 2654435761u
#define PRIME2 805459861u

typedef float v2f __attribute__((ext_vector_type(2)));

__global__ __launch_bounds__(256) void hash_enc_kernel(
    const float* __restrict__ x,      // [N,3]
    const float* __restrict__ bb,     // [6]
    const float* __restrict__ table,  // [16, 1<<19, 2]
    float* __restrict__ enc,          // [N,32]
    float* __restrict__ mask,         // [N]
    int n)
{
    __shared__ __align__(16) float sx[256 * 3];

    const int t    = (int)threadIdx.x;
    const int base = (int)blockIdx.x * 256;
    const int rem  = (n - base < 256) ? (n - base) : 256;  // points in this tile
    const int ndw  = rem * 3;                              // dwords of x to stage

    // ---- Stage the AoS float3 tile into LDS with CDNA5 async global->LDS loads.
    // Coalesced: lane j loads dword j of the tile directly into LDS byte 4*j.
    {
        const float* gsrc = x + (size_t)base * 3;
        const uint32_t lds0 = (uint32_t)(uintptr_t)&sx[0];
#pragma unroll
        for (int k = 0; k < 3; ++k) {
            int j = t + (k << 8);
            if (j < ndw) {
                uint64_t ga = (uint64_t)(uintptr_t)(gsrc + j);
                uint32_t la = lds0 + (uint32_t)(j << 2);
                asm volatile("global_load_async_to_lds_b32 %0, %1, off"
                             :: "v"(la), "v"(ga) : "memory");
            }
        }
        asm volatile("s_wait_asynccnt 0" ::: "memory");
        __syncthreads();
    }

    if (t >= rem) return;

    // stride-3 dword reads: gcd(3,64)=1 -> no LDS bank conflicts
    const float X = sx[t * 3 + 0];
    const float Y = sx[t * 3 + 1];
    const float Z = sx[t * 3 + 2];

    const float bx0 = bb[0], by0 = bb[1], bz0 = bb[2];
    const float xn = (X - bx0) / (bb[3] - bx0);
    const float yn = (Y - by0) / (bb[4] - by0);
    const float zn = (Z - bz0) / (bb[5] - bz0);

    const int p = base + t;
    {
        bool ok = (xn > 0.f) && (xn < 1.f) &&
                  (yn > 0.f) && (yn < 1.f) &&
                  (zn > 0.f) && (zn < 1.f);
        __builtin_nontemporal_store(ok ? 1.0f : 0.0f, &mask[p]);
    }

    float* orow = enc + (size_t)p * (2 * NLEV);
    const v2f* __restrict__ tb = (const v2f*)table;

#pragma unroll 4
    for (int l = 0; l < NLEV; ++l) {
        const float res = (float)(16u << l);   // floor(16 * 2^l) exact

        float px = xn * res, py = yn * res, pz = zn * res;
        float gx = floorf(px), gy = floorf(py), gz = floorf(pz);
        float fx = px - gx, fy = py - gy, fz = pz - gz;

        uint32_t cx = (uint32_t)(int)gx;
        uint32_t cy = (uint32_t)(int)gy;
        uint32_t cz = (uint32_t)(int)gz;

        // (c+1)*P == c*P + P  -> 2 muls + 3 adds for all six axis terms
        uint32_t ax0 = cx,          ax1 = cx + 1u;
        uint32_t ay0 = cy * PRIME1, ay1 = ay0 + PRIME1;
        uint32_t az0 = cz * PRIME2, az1 = az0 + PRIME2;

        const v2f* tl = tb + ((size_t)l << LOG2T);

        // 8 independent b64 gathers: issue all before consuming (MLP)
        v2f f000 = tl[(ax0 ^ ay0 ^ az0) & TMASK];
        v2f f001 = tl[(ax0 ^ ay0 ^ az1) & TMASK];
        v2f f010 = tl[(ax0 ^ ay1 ^ az0) & TMASK];
        v2f f011 = tl[(ax0 ^ ay1 ^ az1) & TMASK];
        v2f f100 = tl[(ax1 ^ ay0 ^ az0) & TMASK];
        v2f f101 = tl[(ax1 ^ ay0 ^ az1) & TMASK];
        v2f f110 = tl[(ax1 ^ ay1 ^ az0) & TMASK];
        v2f f111 = tl[(ax1 ^ ay1 ^ az1) & TMASK];

        float wx1 = fx, wx0 = 1.f - fx;
        float wy1 = fy, wy0 = 1.f - fy;
        float wz1 = fz, wz0 = 1.f - fz;
        float w00 = wy0 * wz0, w01 = wy0 * wz1;
        float w10 = wy1 * wz0, w11 = wy1 * wz1;

        float a0, a1, w;
        w = wx0 * w00; a0 = w * f000.x;           a1 = w * f000.y;
        w = wx0 * w01; a0 = fmaf(w, f001.x, a0);  a1 = fmaf(w, f001.y, a1);
        w = wx0 * w10; a0 = fmaf(w, f010.x, a0);  a1 = fmaf(w, f010.y, a1);
        w = wx0 * w11; a0 = fmaf(w, f011.x, a0);  a1 = fmaf(w, f011.y, a1);
        w = wx1 * w00; a0 = fmaf(w, f100.x, a0);  a1 = fmaf(w, f100.y, a1);
        w = wx1 * w01; a0 = fmaf(w, f101.x, a0);  a1 = fmaf(w, f101.y, a1);
        w = wx1 * w10; a0 = fmaf(w, f110.x, a0);  a1 = fmaf(w, f110.y, a1);
        w = wx1 * w11; a0 = fmaf(w, f111.x, a0);  a1 = fmaf(w, f111.y, a1);

        v2f r; r.x = a0; r.y = a1;
        // NT store: don't let 138MB of streaming output evict the L2-resident table
        __builtin_nontemporal_store(r, (v2f*)(orow + 2 * l));
    }
}

extern "C" void kernel_launch(void* const* d_in, const int* in_sizes, int n_in,
                              void* d_out, int out_size, void* d_ws, size_t ws_size,
                              hipStream_t stream) {
    const float* x     = (const float*)d_in[0];
    const float* bb    = (const float*)d_in[1];
    const float* table = (const float*)d_in[2];

    const int n = in_sizes[0] / 3;

    float* enc  = (float*)d_out;               // [N,32] first
    float* mask = enc + (size_t)n * (2 * NLEV); // then [N]

    const int blocks = (n + 255) / 256;
    hipLaunchKernelGGL(hash_enc_kernel, dim3(blocks), dim3(256), 0, stream,
                       x, bb, table, enc, mask, n);
}